// SimpleGNN_40492951667286
// MI455X (gfx1250) — compile-verified
//
#include <hip/hip_runtime.h>
#include <stdint.h>

// ---------------------------------------------------------------------------
// 2-layer GCN for MI455X (gfx1250, wave32).
//   deg -> dinv -> [GEMM1 wmma f32] -> self-loop init -> edge scatter (atomics)
//   -> [GEMM2 wmma f32, relu fused on A-load] -> self-loop init -> edge scatter
// N = 50000 (divisible by 16), F = H = 128, C = 64, E = 1.6M.
// ---------------------------------------------------------------------------

typedef __attribute__((ext_vector_type(2))) float v2f;
typedef __attribute__((ext_vector_type(8))) float v8f;

#define F_IN  128
#define F_HID 128
#define F_OUT 64

// ---------------- degree / normalization ----------------

__global__ void zero_u32_kernel(uint32_t* __restrict__ p, int n) {
  int i = blockIdx.x * blockDim.x + threadIdx.x;
  if (i < n) p[i] = 0u;
}

__global__ void deg_count_kernel(const int* __restrict__ dst,
                                 uint32_t* __restrict__ deg, int E) {
  int e = blockIdx.x * blockDim.x + threadIdx.x;
  if (e < E) atomicAdd(&deg[dst[e]], 1u);
}

__global__ void deg_finalize_kernel(const uint32_t* __restrict__ deg,
                                    float* __restrict__ dinv, int n) {
  int i = blockIdx.x * blockDim.x + threadIdx.x;
  if (i < n) dinv[i] = rsqrtf((float)deg[i] + 1.0f);  // +1 = self loop
}

// ---------------- WMMA f32 GEMM: D[M x NCOL] = A[M x K] * B[K x NCOL] -------
// One wave per 16x16 output tile, V_WMMA_F32_16X16X4_F32 chained over K.
// A-frag: lanes 0-15 -> M=lane, K={0,1}; lanes 16-31 -> K={2,3}.
// B-frag mirrors (row = kb + v). C/D: vgpr r -> row r + 8*(lane>=16), col lane&15.
// Requires M % 16 == 0 (true here: 50000 = 3125*16) -> unguarded stores.

template <int K, int NCOL, bool RELU>
__global__ void gemm_wmma_kernel(const float* __restrict__ A,
                                 const float* __restrict__ B,
                                 float* __restrict__ D, int M) {
  constexpr int NTILES = NCOL / 16;
  const int tile  = blockIdx.x * (blockDim.x >> 5) + (threadIdx.x >> 5);
  const int mtile = tile / NTILES;
  const int ntile = tile % NTILES;
  if (mtile * 16 >= M) return;                 // wave-uniform branch

  const int lane = threadIdx.x & 31;
  const int half = lane >> 4;                  // 0 or 1
  const int idx  = lane & 15;                  // M (for A) / N (for B,D)
  const int kb   = half * 2;                   // K sub-offset of this lane half

  const float* __restrict__ ap = A + (size_t)(mtile * 16 + idx) * K + kb;
  const float* __restrict__ bp = B + (size_t)kb * NCOL + ntile * 16 + idx;

  v8f acc = {};
#pragma unroll 4
  for (int k = 0; k < K; k += 4) {
    float a0 = ap[k];
    float a1 = ap[k + 1];
    if (RELU) { a0 = fmaxf(a0, 0.0f); a1 = fmaxf(a1, 0.0f); }
    v2f a; a[0] = a0; a[1] = a1;
    v2f b; b[0] = bp[(size_t)k * NCOL]; b[1] = bp[(size_t)(k + 1) * NCOL];
    acc = __builtin_amdgcn_wmma_f32_16x16x4_f32(
        /*neg_a=*/false, a, /*neg_b=*/false, b,
        /*c_mod=*/(short)0, acc, /*reuse_a=*/false, /*reuse_b=*/false);
  }

  // Unconditional coalesced stores: lane idx writes column (ntile*16+idx) of
  // rows mtile*16 + 8*half + r, r = 0..7.
  float* __restrict__ dp =
      D + (size_t)(mtile * 16 + 8 * half) * NCOL + ntile * 16 + idx;
#pragma unroll
  for (int r = 0; r < 8; ++r) dp[(size_t)r * NCOL] = acc[r];
}

// ---------------- aggregation ----------------

// out[i,f] = bias[f] + h[i,f] * dinv[i]^2   (self-loop message + bias)
template <int NCOL>
__global__ void agg_init_kernel(const float* __restrict__ h,
                                const float* __restrict__ dinv,
                                const float* __restrict__ bias,
                                float* __restrict__ out, int n) {
  int i = blockIdx.x * blockDim.x + threadIdx.x;
  if (i < n * NCOL) {
    const int node = i / NCOL;
    const int f    = i % NCOL;
    const float di = dinv[node];
    out[i] = bias[f] + h[i] * di * di;
  }
}

// One wave per edge: gather h[src] row (vectorized loads), scale, scatter-add
// with hardware f32 atomics (destination rows are L2-resident: 25.6MB << 192MB).
template <int NCOL>
__global__ void agg_edges_kernel(const float* __restrict__ h,
                                 const float* __restrict__ dinv,
                                 const int* __restrict__ src,
                                 const int* __restrict__ dst,
                                 float* __restrict__ out, int E) {
  const int e = blockIdx.x * (blockDim.x >> 5) + (threadIdx.x >> 5);
  if (e >= E) return;                          // wave-uniform
  const int lane = threadIdx.x & 31;
  const int s = src[e];
  const int d = dst[e];
  const float norm = dinv[s] * dinv[d];

  constexpr int PER = NCOL / 32;               // 4 (hid) or 2 (out) floats/lane
  const float* __restrict__ hp = h + (size_t)s * NCOL + lane * PER;
  float* __restrict__ op = out + (size_t)d * NCOL + lane * PER;
#pragma unroll
  for (int j = 0; j < PER; ++j)
    unsafeAtomicAdd(op + j, hp[j] * norm);     // global_atomic_add_f32
}

// ---------------- launch ----------------

static inline size_t alignup(size_t x) { return (x + 255) & ~(size_t)255; }

extern "C" void kernel_launch(void* const* d_in, const int* in_sizes, int n_in,
                              void* d_out, int out_size, void* d_ws, size_t ws_size,
                              hipStream_t stream) {
  const float* x  = (const float*)d_in[0];
  const int*   ei = (const int*)d_in[1];       // [2, E] int32
  const float* W1 = (const float*)d_in[2];
  const float* b1 = (const float*)d_in[3];
  const float* W2 = (const float*)d_in[4];
  const float* b2 = (const float*)d_in[5];
  float* out = (float*)d_out;

  const int N = in_sizes[0] / F_IN;            // 50000
  const int E = in_sizes[1] / 2;               // 1600000
  const int* src = ei;
  const int* dst = ei + E;

  // workspace carve-up
  char* ws = (char*)d_ws;
  uint32_t* deg  = (uint32_t*)ws;                         ws += alignup((size_t)N * 4);
  float*    dinv = (float*)ws;                            ws += alignup((size_t)N * 4);
  float*    h1   = (float*)ws;                            ws += alignup((size_t)N * F_HID * 4);
  float*    a1   = (float*)ws;                            // N*F_HID floats
  float*    h3   = h1;                                    // reuse: h1 dead after agg1

  const int T = 256;

  // 1) degree + dinv
  zero_u32_kernel<<<(N + T - 1) / T, T, 0, stream>>>(deg, N);
  deg_count_kernel<<<(E + T - 1) / T, T, 0, stream>>>(dst, deg, E);
  deg_finalize_kernel<<<(N + T - 1) / T, T, 0, stream>>>(deg, dinv, N);

  // 2) h1 = x @ W1   (WMMA f32; 4 waves / block)
  {
    const int tiles = ((N + 15) / 16) * (F_HID / 16);
    gemm_wmma_kernel<F_IN, F_HID, false>
        <<<(tiles + 3) / 4, 128, 0, stream>>>(x, W1, h1, N);
  }

  // 3) a1 = b1 + self-loop + scatter-add over edges
  agg_init_kernel<F_HID><<<((size_t)N * F_HID + T - 1) / T, T, 0, stream>>>(h1, dinv, b1, a1, N);
  agg_edges_kernel<F_HID><<<(E + 7) / 8, T, 0, stream>>>(h1, dinv, src, dst, a1, E);

  // 4) h3 = relu(a1) @ W2   (relu fused into A-load)
  {
    const int tiles = ((N + 15) / 16) * (F_OUT / 16);
    gemm_wmma_kernel<F_HID, F_OUT, true>
        <<<(tiles + 3) / 4, 128, 0, stream>>>(a1, W2, h3, N);
  }

  // 5) out = b2 + self-loop + scatter-add over edges
  agg_init_kernel<F_OUT><<<((size_t)N * F_OUT + T - 1) / T, T, 0, stream>>>(h3, dinv, b2, out, N);
  agg_edges_kernel<F_OUT><<<(E + 7) / 8, T, 0, stream>>>(h3, dinv, src, dst, out, E);
}